// CTCLoss_79233556676765
// MI455X (gfx1250) — compile-verified
//
#include <hip/hip_runtime.h>
#include <hip/hip_bf16.h>
#include <cstdint>
#include <cstddef>

// Problem constants (from reference): N=64, T=64, C=6736, S=16, L=2S+1=33
#define CTC_N     64
#define CTC_T     64
#define CTC_C     6736
#define CTC_S     16
#define CTC_L     33
#define CTC_BLANK 6735
#define CTC_NEG   (-1e30f)
#define CV4       (CTC_C / 4)   // 1684 float4 per row (exact)

typedef __attribute__((ext_vector_type(2))) float v2f;
typedef __attribute__((ext_vector_type(8))) float v8f;

#if defined(__AMDGCN__) && __has_builtin(__builtin_amdgcn_wmma_f32_16x16x4_f32)
#define HAVE_WMMA_F32X4 1
#endif

#if defined(__gfx1250__) && __has_builtin(__builtin_amdgcn_global_load_async_to_lds_b128)
#define USE_ASYNC_LDS 1
// Builtin signature (from hipcc diagnostic): param0 is
// 'int __vector(4) __device__ *'  ==  AS(1) int __attribute__((vector_size(16)))*
typedef int v4i_vs __attribute__((vector_size(16)));
typedef __attribute__((address_space(1))) v4i_vs v4i_global_t;
typedef __attribute__((address_space(3))) v4i_vs v4i_lds_t;
#endif

// Exact-f32 wave(32) sum. WMMA path: A = 16x4 with A[m][0]=p[m], A[m][1]=0,
// A[m][2]=p[m+16], A[m][3]=0 (that's exactly {vgpr0=p, vgpr1=0} per ISA layout),
// B = all-ones 4x16  =>  D[m][n] = p[m] + p[m+16], replicated across n.
// Lanes 0-15 hold D rows 0..7 in c[0..7]; lanes 16-31 hold rows 8..15.
// Sum the 8 D regs, then fold the two lane-halves with one shfl_xor(16).
__device__ __forceinline__ float wave_sum32(float p) {
#ifdef HAVE_WMMA_F32X4
    v2f a; a[0] = p;    a[1] = 0.0f;
    v2f b; b[0] = 1.0f; b[1] = 1.0f;
    v8f c = {};
    c = __builtin_amdgcn_wmma_f32_16x16x4_f32(false, a, false, b,
                                              (short)0, c, false, false);
    float t = ((c[0] + c[1]) + (c[2] + c[3])) + ((c[4] + c[5]) + (c[6] + c[7]));
    t += __shfl_xor(t, 16, 32);
    return t;
#else
    for (int off = 16; off; off >>= 1) p += __shfl_xor(p, off, 32);
    return p;
#endif
}

// ---------------------------------------------------------------------------
// Kernel 1: per-(n,t) row: stage row to LDS (async B128 path), compute
// logZ = max + log(sum exp(x - max)) in exact f32, then emit the 33
// extended-label log-probs  lp_ext[n,t,l] = pred[n,t,ext[l]] - logZ.
// Grid: N*T blocks of 256 threads (8 waves).
// ---------------------------------------------------------------------------
__global__ __launch_bounds__(256) void ctc_row_kernel(
    const float* __restrict__ pred, const int* __restrict__ gt,
    float* __restrict__ lp_ext) {
    __shared__ __align__(16) float buf[CTC_C];   // 26.9 KB row staging
    __shared__ float redmax[8];
    __shared__ float redsum[8];

    const int row  = blockIdx.x;            // row = n*T + t
    const int n    = row / CTC_T;
    const int tid  = threadIdx.x;
    const int lane = tid & 31;
    const int wid  = tid >> 5;

    const float4* __restrict__ g4 = (const float4*)(pred + (size_t)row * CTC_C);
    float4* b4 = (float4*)buf;

    // ---- stage row into LDS ----
    for (int i = tid; i < CV4; i += 256) {
#ifdef USE_ASYNC_LDS
        __builtin_amdgcn_global_load_async_to_lds_b128(
            (v4i_global_t*)(g4 + i), (v4i_lds_t*)(b4 + i), 0, 0);
#else
        b4[i] = g4[i];
#endif
    }
#ifdef USE_ASYNC_LDS
#if __has_builtin(__builtin_amdgcn_s_wait_asynccnt)
    __builtin_amdgcn_s_wait_asynccnt(0);
#else
    asm volatile("s_wait_asynccnt 0" ::: "memory");
#endif
#endif
    __syncthreads();

    // ---- pass 1: block max ----
    float mx = -3.0e38f;
    for (int i = tid; i < CV4; i += 256) {
        float4 v = b4[i];
        mx = fmaxf(mx, fmaxf(fmaxf(v.x, v.y), fmaxf(v.z, v.w)));
    }
    for (int off = 16; off; off >>= 1) mx = fmaxf(mx, __shfl_xor(mx, off, 32));
    if (lane == 0) redmax[wid] = mx;
    __syncthreads();
    float M = redmax[0];
#pragma unroll
    for (int i = 1; i < 8; ++i) M = fmaxf(M, redmax[i]);

    // ---- pass 2: sum of exp(x - M); wave sum via exact-f32 WMMA ----
    float p = 0.0f;
    for (int i = tid; i < CV4; i += 256) {
        float4 v = b4[i];
        p += (__expf(v.x - M) + __expf(v.y - M)) +
             (__expf(v.z - M) + __expf(v.w - M));
    }
    float wsum = wave_sum32(p);              // uniform control flow: EXEC all-1s
    if (lane == 0) redsum[wid] = wsum;
    __syncthreads();
    float S = redsum[0];
#pragma unroll
    for (int i = 1; i < 8; ++i) S += redsum[i];
    const float logZ = M + __logf(S);

    // ---- gather extended-label log-probs ----
    if (tid < CTC_L) {
        const int lbl = (tid & 1) ? gt[n * CTC_S + (tid >> 1)] : CTC_BLANK;
        lp_ext[(size_t)row * CTC_L + tid] = buf[lbl] - logZ;
    }
}

// ---------------------------------------------------------------------------
// Kernel 2: CTC forward (alpha) recursion, one block (64 thr) per batch n.
// ---------------------------------------------------------------------------
__global__ __launch_bounds__(64) void ctc_alpha_kernel(
    const float* __restrict__ lp_ext, const int* __restrict__ gt,
    float* __restrict__ nll) {
    const int n   = blockIdx.x;
    const int tid = threadIdx.x;

    __shared__ float alpha[2 + CTC_L];   // [0..1] = NEG padding
    __shared__ int   ext[CTC_L];
    __shared__ int   allow2[CTC_L];

    const float* __restrict__ lp = lp_ext + (size_t)n * CTC_T * CTC_L;

    if (tid < CTC_L)
        ext[tid] = (tid & 1) ? gt[n * CTC_S + (tid >> 1)] : CTC_BLANK;
    if (tid < 2) alpha[tid] = CTC_NEG;
    __syncthreads();

    if (tid < CTC_L) {
        allow2[tid] = (tid >= 2) && (ext[tid] != CTC_BLANK) &&
                      (ext[tid] != ext[tid - 2]);
        float a = CTC_NEG;
        if (tid == 0) a = lp[0];
        if (tid == 1) a = lp[1];
        alpha[2 + tid] = a;
    }
    __syncthreads();

    for (int t = 1; t < CTC_T; ++t) {
        float a0 = 0.f, a1 = 0.f, a2 = 0.f, lpt = 0.f;
        if (tid < CTC_L) {
            a0  = alpha[2 + tid];
            a1  = alpha[1 + tid];
            a2  = allow2[tid] ? alpha[tid] : CTC_NEG;
            lpt = lp[(size_t)t * CTC_L + tid];
        }
        __syncthreads();
        if (tid < CTC_L) {
            float m   = fmaxf(a0, fmaxf(a1, a2));
            float lse = m + __logf(__expf(a0 - m) + __expf(a1 - m) +
                                   __expf(a2 - m));
            alpha[2 + tid] = lse + lpt;
        }
        __syncthreads();
    }

    if (tid == 0) {
        float x = alpha[2 + CTC_L - 1], y = alpha[2 + CTC_L - 2];
        float m = fmaxf(x, y);
        float lptot = m + __logf(__expf(x - m) + __expf(y - m));
        nll[n] = (-lptot) / (float)CTC_S;       // nll / target_length
    }
}

// ---------------------------------------------------------------------------
// Kernel 3: deterministic serial mean over N (no atomics -> replay-stable).
// ---------------------------------------------------------------------------
__global__ __launch_bounds__(64) void ctc_mean_kernel(
    const float* __restrict__ nll, float* __restrict__ out) {
    if (threadIdx.x == 0) {
        float s = 0.0f;
        for (int i = 0; i < CTC_N; ++i) s += nll[i];
        out[0] = s / (float)CTC_N;               // * LOSS_WEIGHT (=1)
    }
}

extern "C" void kernel_launch(void* const* d_in, const int* in_sizes, int n_in,
                              void* d_out, int out_size, void* d_ws, size_t ws_size,
                              hipStream_t stream) {
    const float* pred = (const float*)d_in[0];   // [N, T, C] f32 logits
    const int*   gt   = (const int*)d_in[1];     // [N, S] i32 labels
    float*       out  = (float*)d_out;           // scalar f32

    float* lp_ext = (float*)d_ws;                            // N*T*L f32 = 540 KB
    float* nll    = lp_ext + (size_t)CTC_N * CTC_T * CTC_L;  // N f32

    ctc_row_kernel<<<dim3(CTC_N * CTC_T), dim3(256), 0, stream>>>(pred, gt, lp_ext);
    ctc_alpha_kernel<<<dim3(CTC_N), dim3(64), 0, stream>>>(lp_ext, gt, nll);
    ctc_mean_kernel<<<dim3(1), dim3(64), 0, stream>>>(nll, out);
}